// SparseWeightNN_38199439130922
// MI455X (gfx1250) — compile-verified
//
#include <hip/hip_runtime.h>

typedef __attribute__((ext_vector_type(2))) float v2f;
typedef __attribute__((ext_vector_type(8))) float v8f;
typedef __attribute__((ext_vector_type(4))) int   v4i;

#ifndef __has_builtin
#define __has_builtin(x) 0
#endif

#if __has_builtin(__builtin_amdgcn_wmma_f32_16x16x4_f32)
#define USE_WMMA_F32 1
#else
#define USE_WMMA_F32 0
#endif

#if __has_builtin(__builtin_amdgcn_global_load_async_to_lds_b128)
#define USE_ASYNC_LDS 1
#else
#define USE_ASYNC_LDS 0
#endif

#if __has_builtin(__builtin_amdgcn_s_wait_asynccnt)
#define WAIT_ASYNC_0() __builtin_amdgcn_s_wait_asynccnt(0)
#else
#define WAIT_ASYNC_0() asm volatile("s_wait_asynccnt 0x0" ::: "memory")
#endif

// builtin signature (from clang diagnostic): (v4i AS1*, v4i AS3*, imm offset, imm cpol)
#define AS_G128(p) ((__attribute__((address_space(1))) v4i*)(p))
#define AS_L128(p) ((__attribute__((address_space(3))) v4i*)(p))

static constexpr int BATCH = 2048;
static constexpr int IN    = 4096;   // K
static constexpr int UNITS = 4096;   // N
static constexpr int NNZ   = 2048;   // nonzeros per input row
static constexpr int KBLK  = 16;     // K block staged in LDS
static constexpr int LDT   = 20;     // A LDS row stride (dwords): conflict-free, 16B-aligned rows
static constexpr int LDBK  = 136;    // B LDS row stride (dwords): 2*136 % 64 == 16 -> disjoint
                                     // bank groups for the two lane-halves; 544B rows (16B mult.)

// ---------------------------------------------------------------------------
// Kernel 1: zero the dense-W workspace (64 MB)
// ---------------------------------------------------------------------------
__global__ void zero_ws(float4* __restrict__ w, int n4) {
    int i      = blockIdx.x * blockDim.x + threadIdx.x;
    int stride = gridDim.x * blockDim.x;
    float4 z = {0.f, 0.f, 0.f, 0.f};
    for (; i < n4; i += stride) w[i] = z;
}

// ---------------------------------------------------------------------------
// Kernel 2: scatter kernel values into dense W: W[row][cols[row][j]] = k[row][j]
// ---------------------------------------------------------------------------
__global__ void scatter_w(const float* __restrict__ kern,
                          const int* __restrict__ cols,
                          float* __restrict__ W) {
    int gid = blockIdx.x * blockDim.x + threadIdx.x;   // < IN*NNZ = 8388608
    int row = gid >> 11;                               // NNZ == 2048
    int c   = cols[gid];
    W[row * UNITS + c] = kern[gid];
}

// ---------------------------------------------------------------------------
// Kernel 3: C = relu(X @ W + bias) via V_WMMA_F32_16X16X4_F32
//   WG: 256 threads = 8 waves, computes 128(M) x 128(N) tile of C
//   Wave (2x4 layout): 64(M) x 32(N) -> 4x2 fragments of 16x16
//   Tiles staged global->LDS with GLOBAL_LOAD_ASYNC_TO_LDS_B128 (no VGPR staging)
// ---------------------------------------------------------------------------
__global__ __launch_bounds__(256) void gemm_relu(
    const float* __restrict__ X,      // [BATCH][IN]
    const float* __restrict__ W,      // [IN][UNITS] dense
    const float* __restrict__ bias,   // [UNITS]
    float* __restrict__ out)          // [BATCH][UNITS]
{
    __shared__ float sA[2][128 * LDT];    // [m 0..127][k 0..KBLK-1] (padded)
#if USE_ASYNC_LDS
    __shared__ float sB[2][KBLK * LDBK];  // [k 0..KBLK-1][n 0..127] (padded)
#else
    __shared__ float sB[2][128 * LDT];    // [n 0..127][k 0..KBLK-1] (padded)
#endif

    const int t     = threadIdx.x;
    const int lane  = t & 31;
    const int wave  = t >> 5;
    const int half  = lane >> 4;     // 0: lanes 0-15, 1: lanes 16-31
    const int l     = lane & 15;
    const int waveM = wave >> 2;     // 0..1  -> M offset waveM*64
    const int waveN = wave & 3;      // 0..3  -> N offset waveN*32
    const int blockN = blockIdx.x * 128;
    const int blockM = blockIdx.y * 128;

    // staging assignment (256 threads cover 128x16 A and 16x128 B per block)
    const int aChunk = (t & 3) * 4;  // k offset within block: 0,4,8,12
    const int aRow   = t >> 2;       // 0..63 (also +64)
    const int bN     = (t & 31) * 4; // n offset within block: 0..124
    const int bK     = t >> 5;       // 0..7 (also +8)

    v8f acc[4][2];
    v8f vzero = {};
#pragma unroll
    for (int mt = 0; mt < 4; ++mt)
#pragma unroll
        for (int nt = 0; nt < 2; ++nt) acc[mt][nt] = vzero;

#if USE_ASYNC_LDS
    // -------- async global->LDS tile movement (ASYNCcnt-tracked) --------
    auto stage = [&](int kb, int buf) {
        const float* ga = X + (blockM + aRow) * IN + kb + aChunk;
        float*       la = &sA[buf][aRow * LDT + aChunk];
        __builtin_amdgcn_global_load_async_to_lds_b128(AS_G128(ga), AS_L128(la), 0, 0);
        __builtin_amdgcn_global_load_async_to_lds_b128(AS_G128(ga + 64 * IN),
                                                       AS_L128(la + 64 * LDT), 0, 0);
        const float* gb = W + (kb + bK) * UNITS + blockN + bN;
        float*       lb = &sB[buf][bK * LDBK + bN];
        __builtin_amdgcn_global_load_async_to_lds_b128(AS_G128(gb), AS_L128(lb), 0, 0);
        __builtin_amdgcn_global_load_async_to_lds_b128(AS_G128(gb + 8 * UNITS),
                                                       AS_L128(lb + 8 * LDBK), 0, 0);
    };
#else
    // -------- fallback: register staging (as round 1) --------
    float4 aReg[2], bReg[2];
    auto gload = [&](int kb) {
        const float* pa = X + (blockM + aRow) * IN + kb + aChunk;
        aReg[0] = *(const float4*)pa;
        aReg[1] = *(const float4*)(pa + 64 * IN);
        const float* pb = W + (kb + bK) * UNITS + blockN + bN;
        bReg[0] = *(const float4*)pb;
        bReg[1] = *(const float4*)(pb + 8 * UNITS);
    };
    auto sstore = [&](int buf) {
        *(float4*)&sA[buf][aRow * LDT + aChunk]        = aReg[0];
        *(float4*)&sA[buf][(aRow + 64) * LDT + aChunk] = aReg[1];
        const float* b0 = (const float*)&bReg[0];
        const float* b1 = (const float*)&bReg[1];
#pragma unroll
        for (int j = 0; j < 4; ++j) {
            sB[buf][(bN + j) * LDT + bK]     = b0[j];   // transpose to [n][k]
            sB[buf][(bN + j) * LDT + bK + 8] = b1[j];
        }
    };
#endif

    auto compute = [&](int buf) {
#pragma unroll
        for (int ks = 0; ks < KBLK; ks += 4) {
            // f32 16x4 fragment layout: lanes 0-15 hold K=ks,ks+1; lanes 16-31 K=ks+2,ks+3
            const int kk = ks + half * 2;
            v2f af[4], bf[2];
#pragma unroll
            for (int mt = 0; mt < 4; ++mt)
                af[mt] = *(const v2f*)&sA[buf][(waveM * 64 + mt * 16 + l) * LDT + kk];
#pragma unroll
            for (int nt = 0; nt < 2; ++nt) {
                const int n = waveN * 32 + nt * 16 + l;
#if USE_ASYNC_LDS
                v2f b;
                b[0] = sB[buf][kk * LDBK + n];          // B[kk][n]
                b[1] = sB[buf][(kk + 1) * LDBK + n];    // B[kk+1][n]
                bf[nt] = b;
#else
                bf[nt] = *(const v2f*)&sB[buf][n * LDT + kk];
#endif
            }
#pragma unroll
            for (int mt = 0; mt < 4; ++mt) {
#pragma unroll
                for (int nt = 0; nt < 2; ++nt) {
#if USE_WMMA_F32
                    acc[mt][nt] = __builtin_amdgcn_wmma_f32_16x16x4_f32(
                        false, af[mt], false, bf[nt],
                        (short)0, acc[mt][nt], false, false);
#else
                    acc[mt][nt][0] += af[mt][0] * bf[nt][0];
#endif
                }
            }
        }
    };

    // -------- prologue --------
#if USE_ASYNC_LDS
    stage(0, 0);
    WAIT_ASYNC_0();
    __syncthreads();
#else
    gload(0);
    sstore(0);
    __syncthreads();
#endif

    int cur = 0;
    for (int kb = 0; kb < IN; kb += KBLK) {
        const bool more = (kb + KBLK) < IN;
#if USE_ASYNC_LDS
        if (more) stage(kb + KBLK, cur ^ 1);   // async fill of next buffer
        compute(cur);
        WAIT_ASYNC_0();                        // our async writes for next buf landed
        __syncthreads();                       // everyone done reading cur + writing nxt
#else
        if (more) gload(kb + KBLK);
        compute(cur);
        if (more) sstore(cur ^ 1);
        __syncthreads();
#endif
        cur ^= 1;
    }

    // epilogue: 16x16 f32 C/D layout: VGPR v, lane-half h -> row = base + h*8 + v, col = base + l
#pragma unroll
    for (int nt = 0; nt < 2; ++nt) {
        const int col = blockN + waveN * 32 + nt * 16 + l;
        const float bv = bias[col];
#pragma unroll
        for (int mt = 0; mt < 4; ++mt) {
#pragma unroll
            for (int v = 0; v < 8; ++v) {
                const int row = blockM + waveM * 64 + mt * 16 + half * 8 + v;
                float r = acc[mt][nt][v] + bv;
                out[row * UNITS + col] = r > 0.f ? r : 0.f;
            }
        }
    }
}

// ---------------------------------------------------------------------------
extern "C" void kernel_launch(void* const* d_in, const int* in_sizes, int n_in,
                              void* d_out, int out_size, void* d_ws, size_t ws_size,
                              hipStream_t stream) {
    const float* x    = (const float*)d_in[0];   // [2048, 4096]
    const float* kern = (const float*)d_in[1];   // [4096*2048]
    const float* bias = (const float*)d_in[2];   // [4096]
    const int*   cols = (const int*)d_in[3];     // [4096, 2048]
    float*       W    = (float*)d_ws;            // 64 MB dense weights
    float*       out  = (float*)d_out;           // [2048, 4096]

    (void)in_sizes; (void)n_in; (void)out_size; (void)ws_size;

    // 1) zero dense W
    const int n4 = (IN * UNITS) / 4;
    hipLaunchKernelGGL(zero_ws, dim3(2048), dim3(256), 0, stream, (float4*)W, n4);

    // 2) scatter nnz into W
    hipLaunchKernelGGL(scatter_w, dim3((IN * NNZ) / 256), dim3(256), 0, stream,
                       kern, cols, W);

    // 3) GEMM + bias + relu
    dim3 grid(UNITS / 128, BATCH / 128);   // 32 x 16
    hipLaunchKernelGGL(gemm_relu, grid, dim3(256), 0, stream, x, W, bias, out);
}